// FlowCPABLinear_23038204576182
// MI455X (gfx1250) — compile-verified
//
#include <hip/hip_runtime.h>

typedef __attribute__((ext_vector_type(2))) float v2f;
typedef __attribute__((ext_vector_type(8))) float v8f;

#define NC 5
#define NSTEPS 32
#define ELEMS_PER_THREAD 8

#define RDLANE_F(f, l) \
  __builtin_bit_cast(float, __builtin_amdgcn_readlane(__builtin_bit_cast(int, (f)), (l)))

// ---------------------------------------------------------------------------
// Kernel A (1 wave): A = B @ theta via V_WMMA_F32_16X16X4_F32, then build the
// per-cell integration table tab[c] = { exp(a_c h), (expm1(a_c h)/a_c)*b_c,
// a_c*h, 0 } in the workspace.
//
// A-operand (16x4 f32) layout: lanes 0-15 hold row M=lane, K={0,1} in the two
// VGPRs; lanes 16-31 hold row M=lane-16, K={2,3}.  B-operand mirrors this for
// the 4x16 matrix; we only populate column N=0 with theta (rest zero).
// ---------------------------------------------------------------------------
__global__ __launch_bounds__(32) void cpab_setup_kernel(
    const float* __restrict__ theta,   // 6
    const float* __restrict__ Bmat,    // 10 x 6 row-major
    float4* __restrict__ tab)          // 5 entries in d_ws
{
  const int lane = threadIdx.x;
  const int m    = lane & 15;
  const bool hi  = lane >= 16;

  // A operand: first WMMA covers K=0..3, second covers K=4..5 (K=6..7 padded 0)
  v2f a0 = {0.f, 0.f}, a1 = {0.f, 0.f};
  if (m < 2 * NC) {
    const float* row = Bmat + m * 6;
    if (!hi) { a0.x = row[0]; a0.y = row[1]; a1.x = row[4]; a1.y = row[5]; }
    else     { a0.x = row[2]; a0.y = row[3]; /* K=6,7 zero pad */ }
  }
  // B operand: column N=0 carries theta; all other columns zero.
  v2f b0 = {0.f, 0.f}, b1 = {0.f, 0.f};
  if (lane == 0)  { b0.x = theta[0]; b0.y = theta[1]; b1.x = theta[4]; b1.y = theta[5]; }
  if (lane == 16) { b0.x = theta[2]; b0.y = theta[3]; }

  v8f acc = {};
  acc = __builtin_amdgcn_wmma_f32_16x16x4_f32(false, a0, false, b0,
                                              (short)0, acc, false, false);
  acc = __builtin_amdgcn_wmma_f32_16x16x4_f32(false, a1, false, b1,
                                              (short)0, acc, false, false);

  // D layout: D[M][0] -> (VGPR M, lane 0) for M<8 ; (VGPR M-8, lane 16) for M>=8
  float Avec[2 * NC];
  Avec[0] = RDLANE_F(acc[0], 0);
  Avec[1] = RDLANE_F(acc[1], 0);
  Avec[2] = RDLANE_F(acc[2], 0);
  Avec[3] = RDLANE_F(acc[3], 0);
  Avec[4] = RDLANE_F(acc[4], 0);
  Avec[5] = RDLANE_F(acc[5], 0);
  Avec[6] = RDLANE_F(acc[6], 0);
  Avec[7] = RDLANE_F(acc[7], 0);
  Avec[8] = RDLANE_F(acc[0], 16);
  Avec[9] = RDLANE_F(acc[1], 16);

  if (lane < NC) {
    const float h  = 1.0f / (float)NSTEPS;
    const float aa = Avec[2 * lane];
    const float bb = Avec[2 * lane + 1];
    const float t  = aa * h;
    const float eah   = expf(t);
    const float ratio = (fabsf(aa) < 1e-6f) ? h : (expm1f(t) / aa);
    tab[lane] = make_float4(eah, ratio * bb, t, 0.0f);
  }
}

// ---------------------------------------------------------------------------
// Kernel B: streaming CPAB warp.  8 elements / thread (two float4 loads) give
// eight independent chains so the per-step ds_load_b128 latency is hidden.
// Per step per element: v_mul, v_cvt_i32, clamp, shift, ds_load_b128, v_fma,
// v_add.  log-det accumulates Sum(a_c*h) instead of a product + final log.
// ---------------------------------------------------------------------------
__global__ __launch_bounds__(256) void cpab_warp_kernel(
    const float* __restrict__ x,
    const float* __restrict__ pa, const float* __restrict__ pb,
    const float* __restrict__ pc, const float* __restrict__ pd,
    const float4* __restrict__ tab,
    float* __restrict__ out, int n)
{
  __shared__ float4 lds_tab[NC];
  if (threadIdx.x < NC) lds_tab[threadIdx.x] = tab[threadIdx.x];
  __syncthreads();

  const float a = pa[0], b = pb[0], c = pc[0], d = pd[0];
  const float logca = logf(c * a);
  const float eps = 1e-7f;

  const int tid  = blockIdx.x * blockDim.x + threadIdx.x;
  const int base = tid * ELEMS_PER_THREAD;
  if (base + ELEMS_PER_THREAD > n) return;   // n is a multiple of 8

  float4 x0 = *(const float4*)(x + base);
  float4 x1 = *(const float4*)(x + base + 4);

  float phi[ELEMS_PER_THREAD], sa[ELEMS_PER_THREAD];
  {
    float xs[ELEMS_PER_THREAD] = {x0.x, x0.y, x0.z, x0.w, x1.x, x1.y, x1.z, x1.w};
#pragma unroll
    for (int j = 0; j < ELEMS_PER_THREAD; ++j) {
      float xc = fminf(fmaxf(xs[j], eps), 1.0f - eps);
      phi[j] = fmaf(a, xc, b);
      sa[j]  = 0.0f;
    }
  }

#pragma unroll 1
  for (int s = 0; s < NSTEPS; ++s) {
#pragma unroll
    for (int j = 0; j < ELEMS_PER_THREAD; ++j) {
      // clamp(trunc(5*phi),0,4) == clamp(floor(5*phi),0,4) after clipping
      int idx = (int)(phi[j] * (float)NC);
      idx = idx < 0 ? 0 : (idx > NC - 1 ? NC - 1 : idx);
      float4 e = lds_tab[idx];                // ds_load_b128
      phi[j] = fmaf(e.x, phi[j], e.y);        // exact in-cell step
      sa[j] += e.z;                           // Sum a_c * h  (== log J)
    }
  }

  float4 z0, z1, l0, l1;
  z0.x = fmaf(c, phi[0], d); z0.y = fmaf(c, phi[1], d);
  z0.z = fmaf(c, phi[2], d); z0.w = fmaf(c, phi[3], d);
  z1.x = fmaf(c, phi[4], d); z1.y = fmaf(c, phi[5], d);
  z1.z = fmaf(c, phi[6], d); z1.w = fmaf(c, phi[7], d);
  l0.x = logca + sa[0]; l0.y = logca + sa[1];
  l0.z = logca + sa[2]; l0.w = logca + sa[3];
  l1.x = logca + sa[4]; l1.y = logca + sa[5];
  l1.z = logca + sa[6]; l1.w = logca + sa[7];

  *(float4*)(out + base)         = z0;
  *(float4*)(out + base + 4)     = z1;
  *(float4*)(out + n + base)     = l0;
  *(float4*)(out + n + base + 4) = l1;
}

// ---------------------------------------------------------------------------
extern "C" void kernel_launch(void* const* d_in, const int* in_sizes, int n_in,
                              void* d_out, int out_size, void* d_ws, size_t ws_size,
                              hipStream_t stream) {
  const float* x     = (const float*)d_in[0];
  const float* theta = (const float*)d_in[1];
  const float* a     = (const float*)d_in[2];
  const float* b     = (const float*)d_in[3];
  const float* c     = (const float*)d_in[4];
  const float* d     = (const float*)d_in[5];
  const float* Bmat  = (const float*)d_in[6];

  float4* tab = (float4*)d_ws;          // 5 * 16 = 80 bytes of scratch
  const int n = in_sizes[0];

  cpab_setup_kernel<<<1, 32, 0, stream>>>(theta, Bmat, tab);

  const int threads = 256;
  const int blocks  = (n / ELEMS_PER_THREAD + threads - 1) / threads;
  cpab_warp_kernel<<<blocks, threads, 0, stream>>>(x, a, b, c, d, tab,
                                                   (float*)d_out, n);
}